// AddEdges_10187662426876
// MI455X (gfx1250) — compile-verified
//
#include <hip/hip_runtime.h>
#include <math.h>

// Outputs (concatenated flat, reference return order):
//   out[0 : E)        = edge_dist
//   out[E : 4E)       = edge_dir, row-major [E, 3]
//
// Inputs per harness convention:
//   d_in[0] = x          (float32, N*3 elements)
//   d_in[1] = edge_index (integer -> const int*, 2*E elements, layout [2, E])

typedef float f32x4 __attribute__((ext_vector_type(4)));
typedef int   i32x4 __attribute__((ext_vector_type(4)));

// -------- vectorized path: 4 edges per thread, requires E % 4 == 0 --------
__global__ __launch_bounds__(256) void edge_geom_vec4(
    const float* __restrict__ x,
    const int*   __restrict__ ei,
    float*       __restrict__ out,
    int E)
{
    const int t  = blockIdx.x * blockDim.x + threadIdx.x;
    const int n4 = E >> 2;
    if (t >= n4) return;

    // Streaming index loads: touched once -> non-temporal (th:TH_LOAD_NT),
    // keeps the 1.2 MB x-table resident in near caches for the gathers.
    i32x4 src = __builtin_nontemporal_load((const i32x4*)ei + t);
    i32x4 dst = __builtin_nontemporal_load((const i32x4*)(ei + E) + t);

    float dist[4];
    float dir[12];
#pragma unroll
    for (int k = 0; k < 4; ++k) {
        // Gathers use default (RT) temporal policy: x fits in L2 (1.2 MB << 192 MB).
        const float* ps = x + 3 * src[k];
        const float* pd = x + 3 * dst[k];
        float rx = ps[0] - pd[0];
        float ry = ps[1] - pd[1];
        float rz = ps[2] - pd[2];
        float dd  = sqrtf(fmaf(rx, rx, fmaf(ry, ry, rz * rz)));
        float inv = 1.0f / (1.0f + dd);
        dist[k]      = dd;
        dir[3*k + 0] = rx * inv;
        dir[3*k + 1] = ry * inv;
        dir[3*k + 2] = rz * inv;
    }

    // Streaming stores: non-temporal b128.
    f32x4 dv = { dist[0], dist[1], dist[2], dist[3] };
    __builtin_nontemporal_store(dv, (f32x4*)out + t);

    f32x4* db = (f32x4*)(out + E) + 3 * t;   // 16B-aligned since E % 4 == 0
    f32x4 a = { dir[0], dir[1], dir[2],  dir[3]  };
    f32x4 b = { dir[4], dir[5], dir[6],  dir[7]  };
    f32x4 c = { dir[8], dir[9], dir[10], dir[11] };
    __builtin_nontemporal_store(a, db + 0);
    __builtin_nontemporal_store(b, db + 1);
    __builtin_nontemporal_store(c, db + 2);
}

// -------- scalar fallback for E % 4 != 0 (not used for E = 6.4M) --------
__global__ __launch_bounds__(256) void edge_geom_scalar(
    const float* __restrict__ x,
    const int*   __restrict__ ei,
    float*       __restrict__ out,
    int E)
{
    const int e = blockIdx.x * blockDim.x + threadIdx.x;
    if (e >= E) return;

    const int s = ei[e];
    const int d = ei[E + e];
    const float* ps = x + 3 * s;
    const float* pd = x + 3 * d;
    float rx = ps[0] - pd[0];
    float ry = ps[1] - pd[1];
    float rz = ps[2] - pd[2];
    float dd  = sqrtf(fmaf(rx, rx, fmaf(ry, ry, rz * rz)));
    float inv = 1.0f / (1.0f + dd);

    out[e]           = dd;
    out[E + 3*e + 0] = rx * inv;
    out[E + 3*e + 1] = ry * inv;
    out[E + 3*e + 2] = rz * inv;
}

extern "C" void kernel_launch(void* const* d_in, const int* in_sizes, int n_in,
                              void* d_out, int out_size, void* d_ws, size_t ws_size,
                              hipStream_t stream) {
    (void)n_in; (void)out_size; (void)d_ws; (void)ws_size;

    const float* x  = (const float*)d_in[0];
    const int*   ei = (const int*)d_in[1];
    float*       out = (float*)d_out;
    const int E = in_sizes[1] / 2;   // edge_index is [2, E]

    if ((E & 3) == 0) {
        const int n4     = E >> 2;
        const int blocks = (n4 + 255) / 256;
        edge_geom_vec4<<<blocks, 256, 0, stream>>>(x, ei, out, E);
    } else {
        const int blocks = (E + 255) / 256;
        edge_geom_scalar<<<blocks, 256, 0, stream>>>(x, ei, out, E);
    }
}